// Model_65652870087281
// MI455X (gfx1250) — compile-verified
//
#include <hip/hip_runtime.h>

#define IN_F   8192
#define OUT_F  8192
#define BATCH  1024

#define ROW_CHUNKS      16
#define ROWS_PER_CHUNK  (OUT_F / ROW_CHUNKS)   // 512

typedef float v2f __attribute__((ext_vector_type(2)));
typedef float v8f __attribute__((ext_vector_type(8)));

// ---------------------------------------------------------------------------
// ws layout (floats):
//   [0 .. 16*8192)          per-chunk colsum partials
//   [16*8192 .. 17*8192)    w_sum  (final column sums of W)
//   [17*8192]               b_sum
// ---------------------------------------------------------------------------
#define WS_PARTIALS 0
#define WS_WSUM     (ROW_CHUNKS * IN_F)
#define WS_BSUM     (WS_WSUM + IN_F)

// --------------------------------------------------------------------------
// Kernel 1: column sums of W via V_WMMA_F32_16X16X4_F32 with A = ones.
// D[0,n] = sum_k B[k,n]; accumulating C across the row loop yields the
// colsum of the whole 512-row chunk for a 16-column strip per wave.
// grid = (64, 16), block = 256  -> 64*8 = 512 strips of 16 cols, 16 row chunks
// --------------------------------------------------------------------------
__global__ __launch_bounds__(256)
void colsum_wmma(const float* __restrict__ W, float* __restrict__ ws) {
    const int lane    = threadIdx.x & 31;
    const int wave    = threadIdx.x >> 5;
    const int laneLow = lane & 15;
    const int rsel    = (lane >> 4) & 1;            // 0: lanes 0-15, 1: lanes 16-31
    const int chunk   = blockIdx.y;
    const int col0    = (blockIdx.x * 8 + wave) * 16;
    const int row0    = chunk * ROWS_PER_CHUNK;

    // vgpr b.x <- row (row0 + 2*rsel), b.y <- row (row0 + 2*rsel + 1)
    // => the 4 rows {k..k+3} of each tile are covered exactly once per column,
    //    so D row 0 = colsum regardless of the hardware's K-slot ordering.
    const float* px = W + (size_t)(row0 + 2 * rsel) * IN_F + col0 + laneLow;

    v8f c = {};                       // f32 16x16 accumulator (8 VGPRs)
    v2f a; a.x = 1.0f; a.y = 1.0f;    // A = ones(16x4)

    #pragma unroll 4
    for (int it = 0; it < ROWS_PER_CHUNK / 4; ++it) {
        v2f b;
        b.x = px[0];
        b.y = px[IN_F];
        // (neg_a, A, neg_b, B, c_mod, C, reuse_a, reuse_b)
        c = __builtin_amdgcn_wmma_f32_16x16x4_f32(
                false, a, false, b, (short)0, c, false, false);
        px += (size_t)4 * IN_F;
    }

    // D VGPR0, lanes 0-15 hold M=0, N=0..15  == the 16 column sums
    if (lane < 16)
        ws[WS_PARTIALS + (size_t)chunk * IN_F + col0 + laneLow] = c[0];
}

// --------------------------------------------------------------------------
// Kernel 2: fold the 16 chunk partials into w_sum (deterministic order).
// grid = 32, block = 256
// --------------------------------------------------------------------------
__global__ __launch_bounds__(256)
void reduce_partials(float* __restrict__ ws) {
    const int i = blockIdx.x * 256 + threadIdx.x;   // 0..8191
    float s = 0.0f;
    #pragma unroll
    for (int ch = 0; ch < ROW_CHUNKS; ++ch)
        s += ws[WS_PARTIALS + (size_t)ch * IN_F + i];
    ws[WS_WSUM + i] = s;
}

// --------------------------------------------------------------------------
// Kernel 3: b_sum = sum(b). One block.
// --------------------------------------------------------------------------
__global__ __launch_bounds__(256)
void bias_sum(const float* __restrict__ b, float* __restrict__ ws) {
    __shared__ float red[8];
    const int tid = threadIdx.x;
    float acc = 0.0f;
    for (int i = tid; i < OUT_F; i += 256) acc += b[i];
    #pragma unroll
    for (int off = 16; off > 0; off >>= 1)
        acc += __shfl_down(acc, off, 32);
    if ((tid & 31) == 0) red[tid >> 5] = acc;
    __syncthreads();
    if (tid == 0) {
        float s = 0.0f;
        #pragma unroll
        for (int w = 0; w < 8; ++w) s += red[w];
        ws[WS_BSUM] = s;
    }
}

// --------------------------------------------------------------------------
// Kernel 4: y[row] = dot(x[row,:], w_sum) + b_sum.  One block per row.
// float4 loads; w_sum (32 KB) stays L2-resident across the 1024 blocks.
// --------------------------------------------------------------------------
__global__ __launch_bounds__(256)
void gemv_rowsum(const float* __restrict__ x, const float* __restrict__ ws,
                 float* __restrict__ out) {
    const int row = blockIdx.x;
    const int tid = threadIdx.x;
    __shared__ float red[8];

    const float4* xr = (const float4*)(x + (size_t)row * IN_F);
    const float4* wv = (const float4*)(ws + WS_WSUM);

    float acc = 0.0f;
    #pragma unroll
    for (int j = 0; j < IN_F / (256 * 4); ++j) {    // 8 iterations
        const float4 xv = xr[j * 256 + tid];
        const float4 w4 = wv[j * 256 + tid];
        acc += xv.x * w4.x + xv.y * w4.y + xv.z * w4.z + xv.w * w4.w;
    }
    #pragma unroll
    for (int off = 16; off > 0; off >>= 1)
        acc += __shfl_down(acc, off, 32);
    if ((tid & 31) == 0) red[tid >> 5] = acc;
    __syncthreads();
    if (tid == 0) {
        float s = ws[WS_BSUM];
        #pragma unroll
        for (int w = 0; w < 8; ++w) s += red[w];
        out[row] = s;
    }
}

// --------------------------------------------------------------------------
extern "C" void kernel_launch(void* const* d_in, const int* in_sizes, int n_in,
                              void* d_out, int out_size, void* d_ws, size_t ws_size,
                              hipStream_t stream) {
    (void)in_sizes; (void)n_in; (void)out_size; (void)ws_size;
    const float* x = (const float*)d_in[0];   // (1024, 8192)
    const float* W = (const float*)d_in[1];   // (8192, 8192)
    const float* b = (const float*)d_in[2];   // (8192,)
    float* out = (float*)d_out;               // (1024,)
    float* ws  = (float*)d_ws;

    // 268 MB of W traffic, fully parallel: 1024 blocks / 8192 waves.
    colsum_wmma   <<<dim3(64, ROW_CHUNKS), 256, 0, stream>>>(W, ws);
    bias_sum      <<<1,                    256, 0, stream>>>(b, ws);
    reduce_partials<<<IN_F / 256,          256, 0, stream>>>(ws);
    gemv_rowsum   <<<BATCH,                256, 0, stream>>>(x, ws, out);
}